// SEGNNLayer_20229295964664
// MI455X (gfx1250) — compile-verified
//
#include <hip/hip_runtime.h>

typedef __attribute__((ext_vector_type(2))) float v2f;
typedef __attribute__((ext_vector_type(8))) float v8f;

__device__ __forceinline__ float fsig(float x) { return 1.0f / (1.0f + __expf(-x)); }
__device__ __forceinline__ float fsilu(float x) { return x * fsig(x); }

// Multi-tile 16x16 f32 GEMM via V_WMMA_F32_16X16X4_F32.
// A: LDS row-major [16][LDA], LDA even -> A fragment = one aligned ds_load_b64.
// BP: LDS K-pair-interleaved weights: BP[(j*LDC + col)*2 + p] = W[2j+p][col].
//     Fragment rows (k+2h, k+2h+1) = pair j = k/2 + h -> one aligned ds_load_b64.
// Fragment layouts per CDNA5 ISA: A lane<16 holds (M=lane, K=k..k+1),
// lane>=16 holds (M=lane-16, K=k+2..k+3); B mirrored; D: c[i] = (M=i+8*half, N=lane&15).
template<int K, int LDA, int LDC, int NT>
__device__ __forceinline__ void gemm16n(const float* A, const float* BP, v8f* acc,
                                        int l, int h, int h2) {
#pragma unroll
  for (int k = 0; k < K; k += 4) {
    v2f a = *(const v2f*)(A + l * LDA + k + h2);
#pragma unroll
    for (int t = 0; t < NT; ++t) {
      v2f b = *(const v2f*)(BP + (((k >> 1) + h) * LDC + t * 16 + l) * 2);
      acc[t] = __builtin_amdgcn_wmma_f32_16x16x4_f32(false, a, false, b, (short)0, acc[t], false, false);
    }
  }
}

__device__ __forceinline__ void load_pad(float* dst, const float* __restrict__ src,
                                         int rows, int cols, int ldd, int tid, int nthr) {
  for (int i = tid; i < rows * ldd; i += nthr) {
    int r = i / ldd, c = i - r * ldd;
    dst[i] = (c < cols) ? src[r * cols + c] : 0.0f;
  }
}

// K-pair-interleave weights into LDS: logical W = [s0(c0 cols) | 0 | s1(c1 cols)@off1 | 0],
// rowsK even. dst[(j*ldc + c)*2 + p] = W[2j+p][c].
__device__ __forceinline__ void load_pairs2(float* dst, const float* __restrict__ s0, int c0,
                                            const float* __restrict__ s1, int c1, int off1,
                                            int rowsK, int ldc, int tid, int nthr) {
  int tot = (rowsK >> 1) * ldc;
  for (int i = tid; i < tot; i += nthr) {
    int j = i / ldc, c = i - j * ldc;
    float v0 = 0.0f, v1 = 0.0f;
    if (c < c0) {
      v0 = s0[(2 * j) * c0 + c];
      v1 = s0[(2 * j + 1) * c0 + c];
    } else if (c >= off1 && c < off1 + c1) {
      int cc = c - off1;
      v0 = s1[(2 * j) * c1 + cc];
      v1 = s1[(2 * j + 1) * c1 + cc];
    }
    dst[i * 2 + 0] = v0;
    dst[i * 2 + 1] = v1;
  }
}

__global__ void segnn_zero(float* __restrict__ p, long n) {
  long i = (long)blockIdx.x * blockDim.x + threadIdx.x;
  if (i < n) p[i] = 0.0f;
}

// -------------------- Edge (message) kernel: m0 -> gate -> m1 -> gate -> atomic scatter
#define EW_PW 3600
__global__ __launch_bounds__(64) void segnn_edge(
    const float* __restrict__ x0, const float* __restrict__ x1,
    const float* __restrict__ ea0, const float* __restrict__ ea1,
    const float* __restrict__ add0,
    const int* __restrict__ snd, const int* __restrict__ rcv,
    const float* __restrict__ m0W0a, const float* __restrict__ m0W0b, const float* __restrict__ m0b0,
    const float* __restrict__ m0W1a, const float* __restrict__ m0W1b,
    const float* __restrict__ m1W0a, const float* __restrict__ m1W0b, const float* __restrict__ m1b0,
    const float* __restrict__ m1W1a, const float* __restrict__ m1W1b,
    float* __restrict__ agg0, float* __restrict__ agg1, int E) {
  // pair-interleaved weights -- shared by both waves of the block
  __shared__ __align__(16) float wBig[72 * 64];   // [W0a(40) | 0 | W1b(8)@48 | 0]
  __shared__ __align__(16) float wXB0[16 * 48];
  __shared__ __align__(16) float wVA[16 * 16];
  __shared__ __align__(16) float zBig[32 * 64];   // m1: [W0a(40) | 0 | W1b(8)@48 | 0]
  __shared__ __align__(16) float zXB0[8 * 48];
  __shared__ __align__(16) float zVA[8 * 16];
  __shared__ __align__(16) float wb0[48], zb0[48];
  __shared__ __align__(16) float stg[2 * EW_PW];  // per-wave staging

  const int tid = threadIdx.x, nthr = blockDim.x;
  load_pairs2(wBig, m0W0a, 40, m0W1b, 8, 48, 72, 64, tid, nthr);
  load_pairs2(wXB0, m0W0b, 40, m0W0b, 0, 48, 16, 48, tid, nthr);
  load_pairs2(wVA, m0W1a, 8, m0W1a, 0, 16, 16, 16, tid, nthr);
  load_pairs2(zBig, m1W0a, 40, m1W1b, 8, 48, 32, 64, tid, nthr);
  load_pairs2(zXB0, m1W0b, 40, m1W0b, 0, 48, 8, 48, tid, nthr);
  load_pairs2(zVA, m1W1a, 8, m1W1a, 0, 16, 8, 16, tid, nthr);
  load_pad(wb0, m0b0, 1, 40, 48, tid, nthr);
  load_pad(zb0, m1b0, 1, 40, 48, tid, nthr);
  __syncthreads();

  const int wave = tid >> 5, lane = tid & 31;
  const int l = lane & 15, h = lane >> 4, h2 = h << 1;
  float* S = stg + wave * EW_PW;
  float* IN0 = S;             // [16][74]  concat(x0[snd], x0[rcv], add0)
  float* X1s = S + 1184;      // 3 x [16][18]  x1cat per spatial dim d
  float* XD  = S + 2048;      // [16][18]  sum_d x1cat*ea1
  float* EA0 = S + 2336;      // [16]
  float* EA1 = S + 2352;      // [16][4]
  float* S1A = S + 2416;      // [16][34]  silu(m0 scalar feats)
  float* V1  = S + 2960;      // 3 x [16][10] gated m0 vectors
  float* XD1 = S + 3440;      // [16][10]

  const long e0 = ((long)blockIdx.x * 2 + wave) * 16;
  const long Emax = (long)E - 1;

  for (int idx = lane; idx < 16 * 72; idx += 32) {
    int r = idx / 72, c = idx - r * 72;
    long e = e0 + r; if (e > Emax) e = Emax;
    float v;
    if (c < 32)      v = x0[(long)snd[e] * 32 + c];
    else if (c < 64) v = x0[(long)rcv[e] * 32 + (c - 32)];
    else             v = add0[e * 8 + (c - 64)];
    IN0[r * 74 + c] = v;
  }
  for (int idx = lane; idx < 16 * 16; idx += 32) {
    int r = idx >> 4, c = idx & 15;
    long e = e0 + r; if (e > Emax) e = Emax;
    long node = (c < 8) ? (long)snd[e] : (long)rcv[e];
    long xb = node * 24 + (long)(c & 7) * 3;
    X1s[0 * 288 + r * 18 + c] = x1[xb + 0];
    X1s[1 * 288 + r * 18 + c] = x1[xb + 1];
    X1s[2 * 288 + r * 18 + c] = x1[xb + 2];
  }
  if (lane < 16) {
    long e = e0 + lane; if (e > Emax) e = Emax;
    EA0[lane] = ea0[e];
    EA1[lane * 4 + 0] = ea1[e * 3 + 0];
    EA1[lane * 4 + 1] = ea1[e * 3 + 1];
    EA1[lane * 4 + 2] = ea1[e * 3 + 2];
  }
  __syncthreads();
  for (int idx = lane; idx < 256; idx += 32) {
    int r = idx >> 4, c = idx & 15;
    XD[r * 18 + c] = X1s[0 * 288 + r * 18 + c] * EA1[r * 4 + 0]
                   + X1s[1 * 288 + r * 18 + c] * EA1[r * 4 + 1]
                   + X1s[2 * 288 + r * 18 + c] * EA1[r * 4 + 2];
  }
  __syncthreads();

  const v8f zero = {0.f, 0.f, 0.f, 0.f, 0.f, 0.f, 0.f, 0.f};

  // ---- m0: one 4-tile GEMM over IN0 gives s-tiles 0..2 (W0a) and tb (W1b)
  v8f acc[4] = {zero, zero, zero, zero};
  gemm16n<72, 74, 64, 4>(IN0, wBig, acc, l, h, h2);
  v8f tb = acc[3];
#pragma unroll
  for (int t = 0; t < 3; ++t)
#pragma unroll
    for (int i = 0; i < 8; ++i) acc[t][i] *= EA0[i + 8 * h];   // a0 scaling of x0-path only
  gemm16n<16, 18, 48, 3>(XD, wXB0, acc, l, h, h2);             // + XD @ W0b
#pragma unroll
  for (int t = 0; t < 3; ++t) {
    float b = wb0[t * 16 + l];
#pragma unroll
    for (int i = 0; i < 8; ++i) acc[t][i] += b;
  }
  // ---- m0 vectors: v_d = ea0 * (X1_d@W1a) + tb * ea1_d
  v8f vf[3];
#pragma unroll
  for (int d = 0; d < 3; ++d) {
    v8f c = zero;
    gemm16n<16, 18, 16, 1>(X1s + d * 288, wVA, &c, l, h, h2);
#pragma unroll
    for (int i = 0; i < 8; ++i) {
      int e = i + 8 * h;
      c[i] = c[i] * EA0[e] + tb[i] * EA1[e * 4 + d];
    }
    vf[d] = c;
  }
  // ---- gate m0, stage m1 inputs (D fragment (M=i+8h, N=l) aligns with gate fragment)
#pragma unroll
  for (int t = 0; t < 2; ++t)
#pragma unroll
    for (int i = 0; i < 8; ++i)
      S1A[(i + 8 * h) * 34 + t * 16 + l] = fsilu(acc[t][i]);
  if (l < 8) {
#pragma unroll
    for (int i = 0; i < 8; ++i) {
      int e = i + 8 * h;
      float g = fsig(acc[2][i]);
      float a = 0.f;
#pragma unroll
      for (int d = 0; d < 3; ++d) {
        float gv = vf[d][i] * g;
        V1[d * 160 + e * 10 + l] = gv;
        a += gv * EA1[e * 4 + d];
      }
      XD1[e * 10 + l] = a;
    }
  }
  __syncthreads();

  // ---- m1: 4-tile GEMM over S1A -> s2 tiles + tb2
  v8f a2[4] = {zero, zero, zero, zero};
  gemm16n<32, 34, 64, 4>(S1A, zBig, a2, l, h, h2);
  v8f tb2 = a2[3];
#pragma unroll
  for (int t = 0; t < 3; ++t)
#pragma unroll
    for (int i = 0; i < 8; ++i) a2[t][i] *= EA0[i + 8 * h];
  gemm16n<8, 10, 48, 3>(XD1, zXB0, a2, l, h, h2);
#pragma unroll
  for (int t = 0; t < 3; ++t) {
    float b = zb0[t * 16 + l];
#pragma unroll
    for (int i = 0; i < 8; ++i) a2[t][i] += b;
  }
  v8f v2[3];
#pragma unroll
  for (int d = 0; d < 3; ++d) {
    v8f c = zero;
    gemm16n<8, 10, 16, 1>(V1 + d * 160, zVA, &c, l, h, h2);
#pragma unroll
    for (int i = 0; i < 8; ++i) {
      int e = i + 8 * h;
      c[i] = c[i] * EA0[e] + tb2[i] * EA1[e * 4 + d];
    }
    v2[d] = c;
  }
  // ---- gate + scatter-add to aggregates
#pragma unroll
  for (int t = 0; t < 2; ++t)
#pragma unroll
    for (int i = 0; i < 8; ++i) {
      long e = e0 + i + 8 * h;
      if (e < (long)E)
        atomicAdd(&agg0[(long)rcv[e] * 32 + t * 16 + l], fsilu(a2[t][i]));
    }
  if (l < 8) {
#pragma unroll
    for (int i = 0; i < 8; ++i) {
      long e = e0 + i + 8 * h;
      if (e < (long)E) {
        float g = fsig(a2[2][i]);
        long rb = (long)rcv[e] * 24 + (long)l * 3;
        atomicAdd(&agg1[rb + 0], v2[0][i] * g);
        atomicAdd(&agg1[rb + 1], v2[1][i] * g);
        atomicAdd(&agg1[rb + 2], v2[2][i] * g);
      }
    }
  }
}

// -------------------- Node (update) kernel: u0 (gated) -> u1 (plain) -> residual + output
#define NW_PW 3472
__global__ __launch_bounds__(64) void segnn_node(
    const float* __restrict__ x0, const float* __restrict__ x1,
    const float* __restrict__ na0, const float* __restrict__ na1,
    const float* __restrict__ u0W0a, const float* __restrict__ u0W0b, const float* __restrict__ u0b0,
    const float* __restrict__ u0W1a, const float* __restrict__ u0W1b,
    const float* __restrict__ u1W0a, const float* __restrict__ u1W0b, const float* __restrict__ u1b0,
    const float* __restrict__ u1W1a, const float* __restrict__ u1W1b,
    const float* __restrict__ agg0, const float* __restrict__ agg1,
    float* __restrict__ out, int N) {
  __shared__ __align__(16) float wBig[64 * 64];   // u0: [W0a(40) | 0 | W1b(8)@48 | 0]
  __shared__ __align__(16) float wXB0[16 * 48];
  __shared__ __align__(16) float wVA[16 * 16];
  __shared__ __align__(16) float zBig[32 * 48];   // u1: [W0a(32) | W1b(8)@32 | 0]
  __shared__ __align__(16) float zXB0[8 * 32];
  __shared__ __align__(16) float zVA[8 * 16];
  __shared__ __align__(16) float wb0[48], zb0[32];
  __shared__ __align__(16) float stg[2 * NW_PW];

  const int tid = threadIdx.x, nthr = blockDim.x;
  load_pairs2(wBig, u0W0a, 40, u0W1b, 8, 48, 64, 64, tid, nthr);
  load_pairs2(wXB0, u0W0b, 40, u0W0b, 0, 48, 16, 48, tid, nthr);
  load_pairs2(wVA, u0W1a, 8, u0W1a, 0, 16, 16, 16, tid, nthr);
  load_pairs2(zBig, u1W0a, 32, u1W1b, 8, 32, 32, 48, tid, nthr);
  load_pairs2(zXB0, u1W0b, 32, u1W0b, 0, 32, 8, 32, tid, nthr);
  load_pairs2(zVA, u1W1a, 8, u1W1a, 0, 16, 8, 16, tid, nthr);
  load_pad(wb0, u0b0, 1, 40, 48, tid, nthr);
  load_pad(zb0, u1b0, 1, 32, 32, tid, nthr);
  __syncthreads();

  const int wave = tid >> 5, lane = tid & 31;
  const int l = lane & 15, h = lane >> 4, h2 = h << 1;
  float* S = stg + wave * NW_PW;
  float* IN0 = S;             // [16][66]  concat(x0, agg0)
  float* X1s = S + 1056;      // 3 x [16][18]  concat(x1, agg1) per d
  float* XD  = S + 1920;      // [16][18]
  float* NA0 = S + 2208;      // [16]
  float* NA1 = S + 2224;      // [16][4]
  float* S1A = S + 2288;      // [16][34]
  float* V1  = S + 2832;      // 3 x [16][10]
  float* XD1 = S + 3312;      // [16][10]

  const long n0 = ((long)blockIdx.x * 2 + wave) * 16;
  const long Nmax = (long)N - 1;

  for (int idx = lane; idx < 16 * 64; idx += 32) {
    int r = idx >> 6, c = idx & 63;
    long n = n0 + r; if (n > Nmax) n = Nmax;
    IN0[r * 66 + c] = (c < 32) ? x0[n * 32 + c] : agg0[n * 32 + (c - 32)];
  }
  for (int idx = lane; idx < 16 * 16; idx += 32) {
    int r = idx >> 4, c = idx & 15;
    long n = n0 + r; if (n > Nmax) n = Nmax;
    const float* src = (c < 8) ? (x1 + n * 24 + (long)c * 3)
                               : (agg1 + n * 24 + (long)(c - 8) * 3);
    X1s[0 * 288 + r * 18 + c] = src[0];
    X1s[1 * 288 + r * 18 + c] = src[1];
    X1s[2 * 288 + r * 18 + c] = src[2];
  }
  if (lane < 16) {
    long n = n0 + lane; if (n > Nmax) n = Nmax;
    NA0[lane] = na0[n];
    NA1[lane * 4 + 0] = na1[n * 3 + 0];
    NA1[lane * 4 + 1] = na1[n * 3 + 1];
    NA1[lane * 4 + 2] = na1[n * 3 + 2];
  }
  __syncthreads();
  for (int idx = lane; idx < 256; idx += 32) {
    int r = idx >> 4, c = idx & 15;
    XD[r * 18 + c] = X1s[0 * 288 + r * 18 + c] * NA1[r * 4 + 0]
                   + X1s[1 * 288 + r * 18 + c] * NA1[r * 4 + 1]
                   + X1s[2 * 288 + r * 18 + c] * NA1[r * 4 + 2];
  }
  __syncthreads();

  const v8f zero = {0.f, 0.f, 0.f, 0.f, 0.f, 0.f, 0.f, 0.f};

  // ---- u0 (gated): 4-tile GEMM over IN0
  v8f acc[4] = {zero, zero, zero, zero};
  gemm16n<64, 66, 64, 4>(IN0, wBig, acc, l, h, h2);
  v8f tb = acc[3];
#pragma unroll
  for (int t = 0; t < 3; ++t)
#pragma unroll
    for (int i = 0; i < 8; ++i) acc[t][i] *= NA0[i + 8 * h];
  gemm16n<16, 18, 48, 3>(XD, wXB0, acc, l, h, h2);
#pragma unroll
  for (int t = 0; t < 3; ++t) {
    float b = wb0[t * 16 + l];
#pragma unroll
    for (int i = 0; i < 8; ++i) acc[t][i] += b;
  }
  v8f vf[3];
#pragma unroll
  for (int d = 0; d < 3; ++d) {
    v8f c = zero;
    gemm16n<16, 18, 16, 1>(X1s + d * 288, wVA, &c, l, h, h2);
#pragma unroll
    for (int i = 0; i < 8; ++i) {
      int e = i + 8 * h;
      c[i] = c[i] * NA0[e] + tb[i] * NA1[e * 4 + d];
    }
    vf[d] = c;
  }
#pragma unroll
  for (int t = 0; t < 2; ++t)
#pragma unroll
    for (int i = 0; i < 8; ++i)
      S1A[(i + 8 * h) * 34 + t * 16 + l] = fsilu(acc[t][i]);
  if (l < 8) {
#pragma unroll
    for (int i = 0; i < 8; ++i) {
      int e = i + 8 * h;
      float g = fsig(acc[2][i]);
      float a = 0.f;
#pragma unroll
      for (int d = 0; d < 3; ++d) {
        float gv = vf[d][i] * g;
        V1[d * 160 + e * 10 + l] = gv;
        a += gv * NA1[e * 4 + d];
      }
      XD1[e * 10 + l] = a;
    }
  }
  __syncthreads();

  // ---- u1 (no gate): 3-tile GEMM (2 scalar tiles + tb3 at cols 32..47)
  v8f a3[3] = {zero, zero, zero};
  gemm16n<32, 34, 48, 3>(S1A, zBig, a3, l, h, h2);
  v8f tb3 = a3[2];
#pragma unroll
  for (int t = 0; t < 2; ++t)
#pragma unroll
    for (int i = 0; i < 8; ++i) a3[t][i] *= NA0[i + 8 * h];
  gemm16n<8, 10, 32, 2>(XD1, zXB0, a3, l, h, h2);
#pragma unroll
  for (int t = 0; t < 2; ++t) {
    float b = zb0[t * 16 + l];
#pragma unroll
    for (int i = 0; i < 8; ++i) a3[t][i] += b;
  }
  v8f v3[3];
#pragma unroll
  for (int d = 0; d < 3; ++d) {
    v8f c = zero;
    gemm16n<8, 10, 16, 1>(V1 + d * 160, zVA, &c, l, h, h2);
#pragma unroll
    for (int i = 0; i < 8; ++i) {
      int e = i + 8 * h;
      c[i] = c[i] * NA0[e] + tb3[i] * NA1[e * 4 + d];
    }
    v3[d] = c;
  }
  // ---- residual + write output [N,56]
#pragma unroll
  for (int t = 0; t < 2; ++t)
#pragma unroll
    for (int i = 0; i < 8; ++i) {
      long n = n0 + i + 8 * h;
      if (n < (long)N)
        out[n * 56 + t * 16 + l] = x0[n * 32 + t * 16 + l] + a3[t][i];
    }
  if (l < 8) {
#pragma unroll
    for (int i = 0; i < 8; ++i) {
      long n = n0 + i + 8 * h;
      if (n < (long)N) {
        long ob = n * 56 + 32 + (long)l * 3;
        long xb = n * 24 + (long)l * 3;
        out[ob + 0] = x1[xb + 0] + v3[0][i];
        out[ob + 1] = x1[xb + 1] + v3[1][i];
        out[ob + 2] = x1[xb + 2] + v3[2][i];
      }
    }
  }
}

extern "C" void kernel_launch(void* const* d_in, const int* in_sizes, int n_in,
                              void* d_out, int out_size, void* d_ws, size_t ws_size,
                              hipStream_t stream) {
  const float* x0   = (const float*)d_in[0];
  const float* x1   = (const float*)d_in[1];
  const float* na0  = (const float*)d_in[2];
  const float* na1  = (const float*)d_in[3];
  const float* ea0  = (const float*)d_in[4];
  const float* ea1  = (const float*)d_in[5];
  const float* add0 = (const float*)d_in[6];
  const int*   snd  = (const int*)d_in[7];
  const int*   rcv  = (const int*)d_in[8];
  const float* W[20];
  for (int i = 0; i < 20; ++i) W[i] = (const float*)d_in[9 + i];
  // order per block: W0a, W0b, b0, W1a, W1b  for m0, m1, u0, u1

  const int N = in_sizes[0] / 32;
  const int E = in_sizes[7];

  float* agg0 = (float*)d_ws;            // [N,32]
  float* agg1 = agg0 + (size_t)N * 32;   // [N,8,3]
  float* out  = (float*)d_out;

  long zn = (long)N * 56;
  segnn_zero<<<(unsigned)((zn + 255) / 256), 256, 0, stream>>>(agg0, zn);

  int etiles = (E + 15) / 16;
  segnn_edge<<<(unsigned)((etiles + 1) / 2), 64, 0, stream>>>(
      x0, x1, ea0, ea1, add0, snd, rcv,
      W[0], W[1], W[2], W[3], W[4],
      W[5], W[6], W[7], W[8], W[9],
      agg0, agg1, E);

  int ntiles = (N + 15) / 16;
  segnn_node<<<(unsigned)((ntiles + 1) / 2), 64, 0, stream>>>(
      x0, x1, na0, na1,
      W[10], W[11], W[12], W[13], W[14],
      W[15], W[16], W[17], W[18], W[19],
      agg0, agg1, out, N);
}